// Kfactor_57552561766963
// MI455X (gfx1250) — compile-verified
//
#include <hip/hip_runtime.h>

// CDNA5 / gfx1250, wave32. fp32 WMMA (V_WMMA_F32_16X16X4_F32) + TDM
// (tensor_load_to_lds, TENSORcnt) double-buffered D streaming.
// Problem is HBM-bound (135 MB vs 68.7 GFLOP) -> native fp32 matrix ops are free.

typedef __attribute__((ext_vector_type(2))) float        v2f;
typedef __attribute__((ext_vector_type(8))) float        v8f;
typedef __attribute__((ext_vector_type(4))) unsigned int v4u;
typedef __attribute__((ext_vector_type(8))) int          v8i;
typedef __attribute__((ext_vector_type(4))) int          v4i;

#define BATCH   131072
#define NCLUS   16
#define DIM     128          // factor dim (d)
#define KF      64           // factor k
#define ROWS    64           // rows (samples) per workgroup
#define NTHR    128          // 4 waves of 32
#define XP      (DIM + 4)    // padded X stride: bank +4/row -> conflict-free A-frags
#define CP      (KF + 4)     // padded C stride
#define DTP     (DIM + 1)    // padded Dt stride: transpose + B-frag reads conflict-free

// Issue one TDM DMA: copy DIM*KF fp32 (32KB, contiguous) from global to LDS.
// D# per CDNA5 ISA: group0 {count=1, lds_addr, global_addr[56:0],
// type=2}; group1 {data_size=4B, tensor_dim0=tile_dim0=8192, dim0_stride=8192};
// groups 2/3 zero (1-D tile). Tracked by TENSORcnt.
// This toolchain declares the 6-arg builtin: (v4u, v8i, v4i, v4i, v8i, i32 cpol).
__device__ __forceinline__ void tdm_load_32kb(const float* gsrc, float* lds_dst) {
  unsigned long long ga = (unsigned long long)(uintptr_t)gsrc;
  unsigned int       la = (unsigned int)(uintptr_t)lds_dst;   // LDS offset = low 32b
  v4u g0;
  g0.x = 1u;                                                  // count=1 (valid D#)
  g0.y = la;                                                  // lds_addr
  g0.z = (unsigned int)(ga & 0xffffffffull);                  // global_addr[31:0]
  g0.w = (unsigned int)((ga >> 32) & 0x01ffffffull) | (2u << 30); // [56:32] | type=2
  v8i g1;
  g1[0] = (int)(2u << 16);             // workgroup_mask=0, data_size=2 (4 bytes)
  g1[1] = (int)(0x2000u << 16);        // tensor_dim0[15:0]=8192 in bits[63:48]
  g1[2] = (int)(1u << 16);             // tensor_dim0[31:16]=0; tensor_dim1=1
  g1[3] = (int)(0x2000u << 16);        // tile_dim0=8192 in bits[127:112]
  g1[4] = 0;                           // tile_dim1=0 (unused), tile_dim2=0
  g1[5] = 8192;                        // tensor_dim0_stride[31:0]
  g1[6] = 0;
  g1[7] = 0;
  v4i z4 = {0, 0, 0, 0};
  v8i z8 = {0, 0, 0, 0, 0, 0, 0, 0};
  __builtin_amdgcn_tensor_load_to_lds(g0, g1, z4, z4, z8, 0);
}

__global__ __launch_bounds__(NTHR) void kfactor_main(
    const float* __restrict__ x, const float* __restrict__ Dg,
    float* __restrict__ recon, float* __restrict__ loss,
    float* __restrict__ label)
{
  __shared__ float Xs[ROWS][XP];     // x tile
  __shared__ float Ds[2][DIM][KF];   // D_n row-major, TDM double buffer
  __shared__ float Dt[KF][DTP];      // D_n transposed (GEMM2 B-fragments)
  __shared__ float Cs[ROWS][CP];     // C staging: D-layout accum -> A-layout frags
  __shared__ float errb[ROWS];       // cross-lane error accumulator (ds_add_f32)
  __shared__ float xn2[ROWS];
  __shared__ float bestE[ROWS];
  __shared__ int   bestL[ROWS];
  __shared__ int   flag[ROWS];

  const int tid  = threadIdx.x;
  const int wave = tid >> 5;         // warpSize == 32 on gfx1250
  const int lane = tid & 31;
  const int half = lane >> 4;        // K-half select for A/B fragments
  const int l15  = lane & 15;
  const int h2   = half * 2;
  const int arow = wave * 16 + l15;  // this lane's A-fragment row
  const size_t row0 = (size_t)blockIdx.x * ROWS;

  // Prefetch cluster 0's D via TDM immediately (one DMA, issued by wave 0 only).
  if (wave == 0) tdm_load_32kb(Dg, &Ds[0][0][0]);

  // ---- load X tile (coalesced global -> padded LDS) ----
  for (int i = tid; i < ROWS * DIM; i += NTHR) {
    int r = i >> 7, c = i & (DIM - 1);
    Xs[r][c] = x[row0 * DIM + i];
  }
  if (tid < ROWS) { bestE[tid] = 3.0e38f; bestL[tid] = 0; }
  __syncthreads();
  if (tid < ROWS) {
    float s = 0.f;
    for (int c = 0; c < DIM; ++c) { float v = Xs[tid][c]; s = fmaf(v, v, s); }
    xn2[tid] = s;
  }

  v8f by[8];                         // best reconstruction, register-resident
  #pragma unroll
  for (int j = 0; j < 8; ++j) by[j] = {};

  for (int n = 0; n < NCLUS; ++n) {
    const int cur = n & 1;
    // Publish Ds[cur]: wave 0 drains TENSORcnt, barrier broadcasts completion.
    if (wave == 0) __builtin_amdgcn_s_wait_tensorcnt(0);
    __syncthreads();
    // Stream next cluster's D into the other buffer while we compute.
    if (wave == 0 && n + 1 < NCLUS)
      tdm_load_32kb(Dg + (size_t)(n + 1) * DIM * KF, &Ds[cur ^ 1][0][0]);

    const float (*Dc)[KF] = Ds[cur];

    // Dt[k][d] = D[d][k], LDS->LDS transpose; stride 129 => both sides conflict-free
    for (int i = tid; i < DIM * KF; i += NTHR) {
      int k = i & (KF - 1), d = i >> 6;
      Dt[k][d] = Dc[d][k];
    }
    if (tid < ROWS) errb[tid] = 0.f;
    __syncthreads();

    // ---- GEMM1: C(16x64 per wave) = X(16x128) * D(128x64), fp32 WMMA ----
    v8f c0 = {}, c1 = {}, c2 = {}, c3 = {};
    for (int k0 = 0; k0 < DIM; k0 += 4) {
      const int ka = k0 + h2;
      v2f a; a.x = Xs[arow][ka]; a.y = Xs[arow][ka + 1];
      v2f b;
      b.x = Dc[ka][l15];      b.y = Dc[ka + 1][l15];
      c0 = __builtin_amdgcn_wmma_f32_16x16x4_f32(false, a, false, b, (short)0, c0, false, false);
      b.x = Dc[ka][16 + l15]; b.y = Dc[ka + 1][16 + l15];
      c1 = __builtin_amdgcn_wmma_f32_16x16x4_f32(false, a, false, b, (short)0, c1, false, false);
      b.x = Dc[ka][32 + l15]; b.y = Dc[ka + 1][32 + l15];
      c2 = __builtin_amdgcn_wmma_f32_16x16x4_f32(false, a, false, b, (short)0, c2, false, false);
      b.x = Dc[ka][48 + l15]; b.y = Dc[ka + 1][48 + l15];
      c3 = __builtin_amdgcn_wmma_f32_16x16x4_f32(false, a, false, b, (short)0, c3, false, false);
    }
    // spill C (own rows only; in-wave DS ordering suffices, no barrier needed)
    #pragma unroll
    for (int v = 0; v < 8; ++v) {
      const int m = wave * 16 + half * 8 + v;
      Cs[m][l15]      = c0[v];
      Cs[m][16 + l15] = c1[v];
      Cs[m][32 + l15] = c2[v];
      Cs[m][48 + l15] = c3[v];
    }

    // ---- GEMM2: Y(16x128 per wave) = C(16x64) * D^T(64x128) ----
    v8f y[8];
    #pragma unroll
    for (int j = 0; j < 8; ++j) y[j] = {};
    for (int k0 = 0; k0 < KF; k0 += 4) {
      const int ka = k0 + h2;
      v2f a; a.x = Cs[arow][ka]; a.y = Cs[arow][ka + 1];
      #pragma unroll
      for (int j = 0; j < 8; ++j) {
        v2f b; b.x = Dt[ka][j * 16 + l15]; b.y = Dt[ka + 1][j * 16 + l15];
        y[j] = __builtin_amdgcn_wmma_f32_16x16x4_f32(false, a, false, b, (short)0, y[j], false, false);
      }
    }

    // ---- per-row err partial: sum(y^2 - 2xy) over this lane's 8 columns,
    //      reduced across lanes with LDS float atomics (ds_add_f32) ----
    #pragma unroll
    for (int v = 0; v < 8; ++v) {
      const int m = wave * 16 + half * 8 + v;
      float p = 0.f;
      #pragma unroll
      for (int j = 0; j < 8; ++j) {
        float yv = y[j][v];
        float xv = Xs[m][j * 16 + l15];
        p = fmaf(yv, yv - 2.0f * xv, p);
      }
      atomicAdd(&errb[m], p);
    }
    __syncthreads();

    if (tid < ROWS) {
      float e = errb[tid] + xn2[tid];      // ||y-x||^2
      if (e < bestE[tid]) { bestE[tid] = e; bestL[tid] = n; flag[tid] = 1; }
      else flag[tid] = 0;
    }
    __syncthreads();

    // conditional per-row select of winning Y, entirely in registers
    #pragma unroll
    for (int v = 0; v < 8; ++v) {
      const int m = wave * 16 + half * 8 + v;
      const bool take = (flag[m] != 0);
      #pragma unroll
      for (int j = 0; j < 8; ++j) by[j][v] = take ? y[j][v] : by[j][v];
    }
  }

  // ---- stage best Y through LDS (x tile is dead now) for coalesced store ----
  __syncthreads();
  #pragma unroll
  for (int v = 0; v < 8; ++v) {
    const int m = wave * 16 + half * 8 + v;
    #pragma unroll
    for (int j = 0; j < 8; ++j) Xs[m][j * 16 + l15] = by[j][v];
  }
  __syncthreads();
  for (int i = tid; i < ROWS * DIM; i += NTHR) {
    int r = i >> 7, c = i & (DIM - 1);
    recon[row0 * DIM + i] = Xs[r][c];
  }
  if (tid < ROWS) label[row0 + tid] = (float)bestL[tid];
  if (tid == 0) {
    float s = 0.f;
    for (int r = 0; r < ROWS; ++r) s += fmaxf(bestE[r], 0.f);
    atomicAdd(loss, s * (1.0f / ((float)BATCH * (float)DIM)));
  }
}

__global__ void kfactor_init(float* loss) { *loss = 0.f; }

extern "C" void kernel_launch(void* const* d_in, const int* in_sizes, int n_in,
                              void* d_out, int out_size, void* d_ws, size_t ws_size,
                              hipStream_t stream) {
  const float* x  = (const float*)d_in[0];   // [BATCH, DIM] fp32
  const float* Dg = (const float*)d_in[1];   // [NCLUS, DIM, KF] fp32
  float* out   = (float*)d_out;
  float* recon = out;                         // BATCH*DIM floats
  float* lossp = out + (size_t)BATCH * DIM;   // 1 float
  float* labp  = lossp + 1;                   // BATCH labels (stored as float values)
  kfactor_init<<<1, 1, 0, stream>>>(lossp);
  kfactor_main<<<BATCH / ROWS, NTHR, 0, stream>>>(x, Dg, recon, lossp, labp);
}